// Involution_19877108646071
// MI455X (gfx1250) — compile-verified
//
#include <hip/hip_runtime.h>
#include <hip/hip_bf16.h>

#if defined(__HIP_DEVICE_COMPILE__)
#if !__has_builtin(__builtin_amdgcn_wmma_f32_16x16x4_f32)
#error "device pass: missing __builtin_amdgcn_wmma_f32_16x16x4_f32"
#endif
#if __has_builtin(__builtin_amdgcn_rcpf)
#define RCPF(x) __builtin_amdgcn_rcpf(x)
#else
#define RCPF(x) (1.0f / (x))
#endif
#if __has_builtin(__builtin_amdgcn_global_load_async_to_lds_b32)
#define HAVE_ASYNC_LDS 1
#endif
#else
#define RCPF(x) (1.0f / (x))
#endif

typedef float v2f __attribute__((ext_vector_type(2)));
typedef float v4f __attribute__((ext_vector_type(4)));
typedef float v8f __attribute__((ext_vector_type(8)));

#if defined(HAVE_ASYNC_LDS)
typedef __attribute__((address_space(1))) int* gas1_t;   // builtin wants int AS1*
typedef __attribute__((address_space(3))) int* las3_t;   // and int AS3*
#define AS1(p) ((gas1_t)(uintptr_t)(p))
#define AS3(p) ((las3_t)(uint32_t)(uintptr_t)(p))
#endif

#define TILE_H 8
#define TILE_W 16
#define HALO_W 22            // TILE_W + 6
#define XROW   24            // halo row pitch in pixels
#define NPIXH  (14 * XROW)   // 336 halo pixels (14 rows x 24)
#define XP     68            // channel pitch (64 padded -> conflict-free b64/b128 LDS loads)
#define KO     196
#define KOP    208           // padded to 13 N-tiles of 16
#define WP     132           // wbuf pixel pitch (128 padded)
#define CIN    64
#define HW     128

__global__ __launch_bounds__(128)
void involution_fused(const float* __restrict__ x,
                      const float* __restrict__ conv_w,
                      const float* __restrict__ bn_gamma,
                      const float* __restrict__ bn_beta,
                      const float* __restrict__ bn_mean,
                      const float* __restrict__ bn_var,
                      float* __restrict__ out)
{
    __shared__ float xs[NPIXH * XP];     //  91392 B : x tile + halo, channel-last [pix][68]
    __shared__ float bmat2[KOP * XP];    //  56576 B : conv_w, [o][68] channel-contiguous
    __shared__ float wbuf[KOP * WP];     // 109824 B : per-pixel kernels, [o][pixel]
    __shared__ float sscale[KOP];
    __shared__ float sshift[KOP];

    const int bx = blockIdx.x;
    const int b  = bx >> 7;                  // 128 tiles per image
    const int rm = bx & 127;
    const int h0 = (rm >> 3) * TILE_H;
    const int w0 = (rm & 7)  * TILE_W;
    const int tid = threadIdx.x;

    // ---- stage x tile (+3 halo, zero padded at borders) into LDS, channel-last ----
    for (int i = tid; i < CIN * NPIXH; i += 128) {
        int c   = i / NPIXH;         // keep global reads coalesced along w
        int rem = i - c * NPIXH;
        int hr  = rem / XROW;
        int wc  = rem - hr * XROW;
        int hh  = h0 - 3 + hr;
        int ww  = w0 - 3 + wc;
        bool valid = (wc < HALO_W) && ((unsigned)hh < HW) && ((unsigned)ww < HW);
        float* dst = &xs[rem * XP + c];      // swizzle to [pix][c]
#if defined(HAVE_ASYNC_LDS)
        if (valid) {
            // DMA path: global -> LDS without VGPR round-trip (ASYNCcnt-tracked)
            __builtin_amdgcn_global_load_async_to_lds_b32(
                AS1(&x[((b * CIN + c) * HW + hh) * HW + ww]), AS3(dst), 0, 0);
        } else {
            *dst = 0.f;  // disjoint addresses from async writes
        }
#else
        float v = 0.f;
        if (valid)
            v = x[((b * CIN + c) * HW + hh) * HW + ww];
        *dst = v;
#endif
    }
    // ---- stage conv_w as [o][68] channel-contiguous (coalesced reads), zero o>=196 ----
    for (int i = tid; i < KOP * CIN; i += 128) {
        int o = i >> 6;          // i / 64
        int c = i & 63;
        bmat2[o * XP + c] = (o < KO) ? conv_w[i] : 0.f;
    }
    // ---- fold BN (eval) into per-channel affine ----
    if (tid < KOP) {
        float sc = 0.f, sh = 0.f;
        if (tid < KO) {
            sc = bn_gamma[tid] * rsqrtf(bn_var[tid] + 1e-5f);
            sh = bn_beta[tid] - bn_mean[tid] * sc;
        }
        sscale[tid] = sc;
        sshift[tid] = sh;
    }
#if defined(HAVE_ASYNC_LDS)
#if __has_builtin(__builtin_amdgcn_s_wait_asynccnt)
    __builtin_amdgcn_s_wait_asynccnt(0);
#else
    asm volatile("s_wait_asynccnt 0" ::: "memory");
#endif
#endif
    __syncthreads();

    // ---- phase 1: per-pixel kernel generation ----
    // GEMM [128 px x 64 ch] x [64 ch x 208] via V_WMMA_F32_16X16X4_F32,
    // fused BN + SiLU, result kept in LDS wbuf[o][pixel].
    const int wave = tid >> 5;
    const int lane = tid & 31;
    const int mcol = lane & 15;   // M index (pixel column) / N index (o)
    const int hi   = lane >> 4;   // K-pair select per ISA 16x4 f32 layout

    for (int m = wave; m < TILE_H; m += 4) {     // M-tile == tile row
        v2f a[16];                                // all K fragments in regs
        const int abase = ((m + 3) * XROW + 3 + mcol) * XP;  // center pixel, channel 0
#pragma unroll
        for (int kk = 0; kk < 16; ++kk) {
            int c0 = kk * 4 + 2 * hi;            // even -> 8B-aligned ds_load_b64
            a[kk] = *(const v2f*)&xs[abase + c0];
        }
        for (int n = 0; n < 13; ++n) {
            const int bbase = (n * 16 + mcol) * XP + 2 * hi;
            v8f acc = {};
#pragma unroll
            for (int kk = 0; kk < 16; ++kk) {
                // paired channels contiguous -> single ds_load_b64, no re-pairing moves
                v2f bf = *(const v2f*)&bmat2[bbase + kk * 4];
                acc = __builtin_amdgcn_wmma_f32_16x16x4_f32(
                          false, a[kk], false, bf, (short)0, acc, false, false);
            }
            // per C/D layout: lane's 8 values share one o = n*16 + (lane&15)
            const int   o  = n * 16 + mcol;
            const float sc = sscale[o];
            const float sh = sshift[o];
            const int pbase = m * 16 + 8 * hi;   // M = r + 8*(lane>=16)
#pragma unroll
            for (int r = 0; r < 8; ++r) {
                float z  = acc[r] * sc + sh;
                float sg = RCPF(1.0f + __expf(-z));  // fast sigmoid (v_rcp_f32)
                wbuf[o * WP + pbase + r] = z * sg;   // SiLU
            }
        }
    }
    __syncthreads();

    // ---- phase 2: involution gather (49-tap weighted sum per group) ----
    // channel-last layout: 16 group channels contiguous -> ds_load_b128 + v_pk_fma_f32
    const int p  = tid;           // pixel within tile
    const int pr = p >> 4;
    const int pc = p & 15;
    const int h  = h0 + pr;
    const int w  = w0 + pc;
    const int pixbase = pr * XROW + pc;
    for (int g = 0; g < 4; ++g) {
        float wk[49];
#pragma unroll
        for (int k = 0; k < 49; ++k)
            wk[k] = wbuf[(g * 49 + k) * WP + p];
        v4f a0 = {}, a1 = {}, a2 = {}, a3 = {};
#pragma unroll
        for (int dh = 0; dh < 7; ++dh) {
#pragma unroll
            for (int dw = 0; dw < 7; ++dw) {
                const float wv = wk[dh * 7 + dw];
                const v4f wv4 = {wv, wv, wv, wv};
                const float* xp = &xs[(pixbase + dh * XROW + dw) * XP + g * 16];
                a0 = __builtin_elementwise_fma(wv4, *(const v4f*)(xp + 0),  a0);
                a1 = __builtin_elementwise_fma(wv4, *(const v4f*)(xp + 4),  a1);
                a2 = __builtin_elementwise_fma(wv4, *(const v4f*)(xp + 8),  a2);
                a3 = __builtin_elementwise_fma(wv4, *(const v4f*)(xp + 12), a3);
            }
        }
        float* op = &out[((b * CIN + g * 16) * HW + h) * HW + w];
#pragma unroll
        for (int e = 0; e < 4; ++e) {
            op[(e +  0) * HW * HW] = a0[e];
            op[(e +  4) * HW * HW] = a1[e];
            op[(e +  8) * HW * HW] = a2[e];
            op[(e + 12) * HW * HW] = a3[e];
        }
    }
}

extern "C" void kernel_launch(void* const* d_in, const int* in_sizes, int n_in,
                              void* d_out, int out_size, void* d_ws, size_t ws_size,
                              hipStream_t stream) {
    const float* x      = (const float*)d_in[0];
    const float* conv_w = (const float*)d_in[1];
    const float* gamma  = (const float*)d_in[2];
    const float* beta   = (const float*)d_in[3];
    const float* mean   = (const float*)d_in[4];
    const float* var    = (const float*)d_in[5];
    float* out = (float*)d_out;

    dim3 grid(4 * (HW / TILE_H) * (HW / TILE_W));  // 512 blocks
    involution_fused<<<grid, 128, 0, stream>>>(x, conv_w, gamma, beta, mean, var, out);
}